// AutoencoderAdaptGenerative_48249662603936
// MI455X (gfx1250) — compile-verified
//
#include <hip/hip_runtime.h>
#include <hip/hip_bf16.h>
#include <math.h>

typedef __attribute__((ext_vector_type(2))) float v2f;
typedef __attribute__((ext_vector_type(8))) float v8f;

#define LOG2PI 1.8378770664093453f   // n_dim * 0.5 * log(2*pi) with n_dim=2
#define LOG2E  1.4426950408889634f
#define LN2    0.6931471805599453f
#define TILES_PER_WAVE 4

// Transposed mapping (log2 domain): D2[jc, n] = LOG2E * (A[jc,0:4].B[0:4,n] + K[jc])
//   A (16x4, params, pre-scaled by LOG2E), B (4x16, samples): [y0^2, y1^2, y0, y1]
//   K folds -0.5*sum(mu^2/s^2) - sum(log s) - log2pi + log_softmax(pi) + log prior,
//   pre-scaled by LOG2E so the WMMA emits base-2 logits directly.
// D layout puts jc = t*16 + hi*8 + i in VGPR i of lane half hi -> all 8 components
// of symbol j = 2t+hi sit in ONE lane's 8 VGPRs: logsumexp is shuffle-free, and
// in base-2: raw v_exp_f32 / v_log_f32, no per-element log2e muls, no log guards
// (sum of exp2 is in [1,8] since the max term contributes exactly 1).
// Only FULL 16-row tiles here: no clamps, no store guards, EXEC stays all-1s.
__global__ __launch_bounds__(256) void mdn_posterior_wmma(
    const float* __restrict__ y,
    const float* __restrict__ mus,
    const float* __restrict__ sigmas,
    const float* __restrict__ pi_logits,
    const float* __restrict__ prior,
    float* __restrict__ out,
    int n_full_tiles)
{
    __shared__ float W[5 * 128];   // 4 A-param rows (4x128) + K (128), log2-scaled

    const int tid = threadIdx.x;
    if (tid < 128) {
        const int j = tid >> 3;            // symbol
        const int c = tid & 7;             // component
        const int base = j * 16 + c * 2;   // mus/sigmas flat [J, C*D]
        const float m0 = mus[base],    m1 = mus[base + 1];
        const float s0 = sigmas[base], s1 = sigmas[base + 1];
        const float iv0 = 1.0f / (s0 * s0);
        const float iv1 = 1.0f / (s1 * s1);
        W[      tid] = LOG2E * -0.5f * iv0;
        W[128 + tid] = LOG2E * -0.5f * iv1;
        W[256 + tid] = LOG2E * m0 * iv0;
        W[384 + tid] = LOG2E * m1 * iv1;
        float mx = pi_logits[j * 8];
        #pragma unroll
        for (int i = 1; i < 8; ++i) mx = fmaxf(mx, pi_logits[j * 8 + i]);
        float se = 0.0f;
        #pragma unroll
        for (int i = 0; i < 8; ++i) se += __expf(pi_logits[j * 8 + i] - mx);
        const float lse = mx + __logf(se);
        W[512 + tid] = LOG2E * (-0.5f * (m0 * m0 * iv0 + m1 * m1 * iv1)
                                - (__logf(s0) + __logf(s1)) - LOG2PI
                                + (pi_logits[j * 8 + c] - lse) + __logf(prior[j]));
    }
    __syncthreads();

    const int lane = tid & 31;
    const int wave = tid >> 5;
    const int m    = lane & 15;
    const int hi   = lane >> 4;

    const int tile0 = (blockIdx.x * (blockDim.x >> 5) + wave) * TILES_PER_WAVE;
    if (tile0 >= n_full_tiles) return;     // whole-wave exit: EXEC stays all-1s

    // ---- Wave-invariant WMMA operands, hoisted out of the sample loop ----
    v2f a[8];
    v8f kvec[8];
    #pragma unroll
    for (int t = 0; t < 8; ++t) {
        const int ncol = t * 16 + m;
        a[t].x = W[(hi ? 256 :   0) + ncol];
        a[t].y = W[(hi ? 384 : 128) + ncol];
        const float* Kp = &W[512 + t * 16 + hi * 8];   // contiguous -> b128 loads
        #pragma unroll
        for (int i = 0; i < 8; ++i) kvec[t][i] = Kp[i];
    }

    // Strength-reduced pointers: y advances 16 rows (32 floats), out 256 floats.
    const int row0 = tile0 * 16 + m;
    const float* yp = y + row0 * 2;
    float*       op = out + row0 * 16 + hi;

    int count = n_full_tiles - tile0;
    if (count > TILES_PER_WAVE) count = TILES_PER_WAVE;

    #pragma unroll 1
    for (int it = 0; it < count; ++it) {
        const v2f yv = *(const v2f*)yp;
        yp += 32;

        // B (4x16): VGPR0 = {K0=y0^2 | K2=y0}, VGPR1 = {K1=y1^2 | K3=y1}.
        v2f b;
        b.x = hi ? yv.x : yv.x * yv.x;
        b.y = hi ? yv.y : yv.y * yv.y;

        #pragma unroll
        for (int t = 0; t < 8; ++t) {
            const v8f d = __builtin_amdgcn_wmma_f32_16x16x4_f32(
                false, a[t], false, b, (short)0, kvec[t], false, false);

            // In-lane base-2 logsumexp over the 8 components (d[0..7]).
            float mx = fmaxf(fmaxf(d[0], d[1]), fmaxf(d[2], d[3]));
            mx = fmaxf(mx, fmaxf(fmaxf(d[4], d[5]), fmaxf(d[6], d[7])));
            // Raw v_exp_f32: args <= 0, underflow-to-zero is fine.
            float ex0 = __builtin_amdgcn_exp2f(d[0] - mx);
            float ex1 = __builtin_amdgcn_exp2f(d[1] - mx);
            float ex2 = __builtin_amdgcn_exp2f(d[2] - mx);
            float ex3 = __builtin_amdgcn_exp2f(d[3] - mx);
            float ex4 = __builtin_amdgcn_exp2f(d[4] - mx);
            float ex5 = __builtin_amdgcn_exp2f(d[5] - mx);
            float ex6 = __builtin_amdgcn_exp2f(d[6] - mx);
            float ex7 = __builtin_amdgcn_exp2f(d[7] - mx);
            const float e = ((ex0 + ex1) + (ex2 + ex3)) + ((ex4 + ex5) + (ex6 + ex7));
            // e in [1,8]: raw v_log_f32 needs no denorm guard.
            op[2 * t] = LN2 * (mx + __builtin_amdgcn_logf(e));
        }
        op += 256;
    }
}

// Tail kernel: scalar per-(row, j) evaluation for the n_samp % 16 remainder
// rows. Launched only when rem > 0 (never for N = 500000).
__global__ __launch_bounds__(256) void mdn_posterior_tail(
    const float* __restrict__ y,
    const float* __restrict__ mus,
    const float* __restrict__ sigmas,
    const float* __restrict__ pi_logits,
    const float* __restrict__ prior,
    float* __restrict__ out,
    int n_samp, int row0)
{
    const int w = blockIdx.x * blockDim.x + threadIdx.x;   // (row-row0)*16 + j
    if (w >= (n_samp - row0) * 16) return;
    const int row = row0 + (w >> 4);
    const int j   = w & 15;
    const float y0 = y[row * 2], y1 = y[row * 2 + 1];

    float mxp = pi_logits[j * 8];
    #pragma unroll
    for (int i = 1; i < 8; ++i) mxp = fmaxf(mxp, pi_logits[j * 8 + i]);
    float se = 0.0f;
    #pragma unroll
    for (int i = 0; i < 8; ++i) se += __expf(pi_logits[j * 8 + i] - mxp);
    const float lse = mxp + __logf(se);
    const float lp  = __logf(prior[j]);

    float vals[8];
    float vmax = -INFINITY;
    #pragma unroll
    for (int c = 0; c < 8; ++c) {
        const int base = j * 16 + c * 2;
        const float m0 = mus[base],    m1 = mus[base + 1];
        const float s0 = sigmas[base], s1 = sigmas[base + 1];
        const float d0 = (y0 - m0) / s0, d1 = (y1 - m1) / s1;
        const float v = -0.5f * (d0 * d0 + d1 * d1)
                        - (__logf(s0) + __logf(s1)) - LOG2PI
                        + (pi_logits[j * 8 + c] - lse) + lp;
        vals[c] = v;
        vmax = fmaxf(vmax, v);
    }
    float e = 0.0f;
    #pragma unroll
    for (int c = 0; c < 8; ++c) e += __expf(vals[c] - vmax);
    out[row * 16 + j] = vmax + __logf(e);
}

extern "C" void kernel_launch(void* const* d_in, const int* in_sizes, int n_in,
                              void* d_out, int out_size, void* d_ws, size_t ws_size,
                              hipStream_t stream) {
    const float* y         = (const float*)d_in[0];
    const float* mus       = (const float*)d_in[1];
    const float* sigmas    = (const float*)d_in[2];
    const float* pi_logits = (const float*)d_in[3];
    const float* prior     = (const float*)d_in[4];
    float* out = (float*)d_out;

    const int n_samp       = in_sizes[0] / 2;              // y is [N, 2]
    const int n_full_tiles = n_samp / 16;                  // 31250 for N=500000
    const int rem          = n_samp - n_full_tiles * 16;   // 0 for N=500000

    if (n_full_tiles > 0) {
        const int n_super = (n_full_tiles + TILES_PER_WAVE - 1) / TILES_PER_WAVE;
        const int waves_per_block = 8;                     // 256 threads = 8 wave32
        dim3 block(32 * waves_per_block);
        dim3 grid((n_super + waves_per_block - 1) / waves_per_block);
        mdn_posterior_wmma<<<grid, block, 0, stream>>>(y, mus, sigmas, pi_logits,
                                                       prior, out, n_full_tiles);
    }
    if (rem > 0) {
        mdn_posterior_tail<<<1, 256, 0, stream>>>(y, mus, sigmas, pi_logits, prior,
                                                  out, n_samp, n_full_tiles * 16);
    }
}